// MultiBoxLoss_15126874817009
// MI455X (gfx1250) — compile-verified
//
#include <hip/hip_runtime.h>
#include <hip/hip_bf16.h>
#include <math.h>

// Problem constants (match reference)
#define BB   128
#define PP   8732
#define CC   21
#define NOBJ 16
#define JTHRESH 0.5f
#define EPS_LS  0.05f
#define VAR0    0.1f
#define VAR1    0.2f

// Tiling for the async conf_data pipeline
#define NT          ((PP + 255) / 256)          // 35 tiles of 256 priors
#define TILE_BYTES  (256 * CC * 4)              // 21504 B per tile
#define MAIN_PT     6                           // b128 async loads / thread / tile
#define MAIN_BUFW   (MAIN_PT * 256 * 4)         // 6144 words (24576 B, padded)
// k_select row staging
#define SEL_PT      9                           // ceil(2183 chunks / 256)
#define SEL_BUFW    (SEL_PT * 256 * 4)          // 9216 words (36864 B)

typedef float v2f __attribute__((ext_vector_type(2)));
typedef float v8f __attribute__((ext_vector_type(8)));

#define S_WAIT_ASYNC_(n) asm volatile("s_wait_asynccnt " #n ::: "memory")
#define S_WAIT_ASYNC(n)  S_WAIT_ASYNC_(n)

// ---------------------------------------------------------------------------
// Async global->LDS b128 copy (CDNA5 async path, tracked by ASYNCcnt).
// Every thread issues exactly PER_THREAD instructions (uniform per wave) so a
// counted s_wait_asynccnt is exact; out-of-range chunks are clamped to the
// last valid 16B of the source buffer (duplicates land in the LDS pad).
// ---------------------------------------------------------------------------
template <int PER_THREAD>
__device__ __forceinline__ void asyncCopyB128(unsigned lds_base,
                                              const char* gbase,
                                              size_t byte_base,
                                              size_t total_bytes) {
  const unsigned tid = threadIdx.x;
  size_t remain = total_bytes - byte_base;
  unsigned max_idx = (unsigned)((remain >> 4) - 1u);
#pragma unroll
  for (int j = 0; j < PER_THREAD; ++j) {
    unsigned idx = tid + (unsigned)(j * 256);
    unsigned g   = idx > max_idx ? max_idx : idx;
    unsigned long long ga =
        (unsigned long long)(gbase + byte_base + ((size_t)g << 4));
    unsigned lo = lds_base + (idx << 4);
    asm volatile("global_load_async_to_lds_b128 %0, %1, off"
                 :: "v"(lo), "v"(ga) : "memory");
  }
}

// ---------------------------------------------------------------------------
// Wave-level f32 reduction on the matrix core:
// D = A(16x4 f32) * ones(4x16): D[m][n] = x[m] + x[m+16]; summing the 8
// accumulator VGPRs + one shfl_xor(16) yields the full wave sum in f32.
// ---------------------------------------------------------------------------
__device__ __forceinline__ float waveReduceWMMA(float x) {
  v2f a;   a[0] = x;      a[1] = 0.0f;
  v2f one; one[0] = 1.0f; one[1] = 1.0f;
  v8f c = {};
  v8f d = __builtin_amdgcn_wmma_f32_16x16x4_f32(
      false, a, false, one, (short)0, c, false, false);
  float s = d[0] + d[1] + d[2] + d[3] + d[4] + d[5] + d[6] + d[7];
  s += __shfl_xor(s, 16, 32);
  return s;
}

__device__ __forceinline__ float blockReduceF32(float x, float* s8) {
  float w = waveReduceWMMA(x);
  int lane = threadIdx.x & 31;
  int wid  = threadIdx.x >> 5;
  if (lane == 0) s8[wid] = w;
  __syncthreads();
  int nw = (blockDim.x + 31) >> 5;
  float t = 0.0f;
  for (int i = 0; i < nw; ++i) t += s8[i];
  __syncthreads();
  return t;
}

// ---------------------------------------------------------------------------
// Kernel 0: zero accumulators. acc[0]=sl1, acc[1]=ce(pos), acc[2]=ce(neg),
// acc[3]=selected-negative count.
// ---------------------------------------------------------------------------
__global__ void k_init(float* __restrict__ acc, int* __restrict__ num_pos) {
  int t = threadIdx.x;
  if (t < 4)  acc[t] = 0.0f;
  if (t < BB) num_pos[t] = 0;
}

// ---------------------------------------------------------------------------
// Kernel 1: per-truth best prior (argmax over P, first-index tie break)
// packed key = (float_bits(iou) << 32) | (0xFFFFFFFF - p).
// ---------------------------------------------------------------------------
__global__ void k_bestprior(const float* __restrict__ dbox,
                            const float* __restrict__ tboxes,
                            unsigned long long* __restrict__ best_prior) {
  const int b   = blockIdx.x;
  const int tid = threadIdx.x;
  __shared__ float s_t[NOBJ][4];
  __shared__ float s_ta[NOBJ];
  __shared__ unsigned long long s_best[NOBJ];

  if (tid < NOBJ * 4) ((float*)s_t)[tid] = tboxes[b * NOBJ * 4 + tid];
  if (tid < NOBJ)     s_best[tid] = 0ull;
  __syncthreads();
  if (tid < NOBJ)
    s_ta[tid] = (s_t[tid][2] - s_t[tid][0]) * (s_t[tid][3] - s_t[tid][1]);
  __syncthreads();

  unsigned long long m[NOBJ];
#pragma unroll
  for (int j = 0; j < NOBJ; ++j) m[j] = 0ull;

  for (int p = tid; p < PP; p += blockDim.x) {
    float4 db = ((const float4*)dbox)[p];
    float bx0 = db.x - 0.5f * db.z, by0 = db.y - 0.5f * db.w;
    float bx1 = db.x + 0.5f * db.z, by1 = db.y + 0.5f * db.w;
    float ab = db.z * db.w;
    unsigned int pinv = 0xFFFFFFFFu - (unsigned int)p;
#pragma unroll
    for (int j = 0; j < NOBJ; ++j) {
      float ix0 = fmaxf(s_t[j][0], bx0);
      float iy0 = fmaxf(s_t[j][1], by0);
      float ix1 = fminf(s_t[j][2], bx1);
      float iy1 = fminf(s_t[j][3], by1);
      float iw = fmaxf(ix1 - ix0, 0.0f);
      float ih = fmaxf(iy1 - iy0, 0.0f);
      float inter = iw * ih;
      float iou = inter / (s_ta[j] + ab - inter);
      unsigned long long pk =
          ((unsigned long long)__float_as_uint(iou) << 32) | (unsigned long long)pinv;
      m[j] = pk > m[j] ? pk : m[j];
    }
  }
#pragma unroll
  for (int j = 0; j < NOBJ; ++j) atomicMax(&s_best[j], m[j]);
  __syncthreads();
  if (tid < NOBJ) best_prior[b * NOBJ + tid] = s_best[tid];
}

// ---------------------------------------------------------------------------
// Kernel 2: match + smooth-L1 + smoothed CE. conf_data is streamed tile-by-
// tile through LDS with double-buffered global_load_async_to_lds_b128.
// grid = (4, B); block bx handles tiles bx, bx+4, ...
// ---------------------------------------------------------------------------
__global__ void k_main(const float* __restrict__ loc,
                       const float* __restrict__ conf,
                       const float* __restrict__ dbox,
                       const float* __restrict__ tboxes,
                       const int*   __restrict__ tlabels,
                       const unsigned long long* __restrict__ best_prior,
                       float* __restrict__ ce_rank,
                       int*   __restrict__ num_pos,
                       float* __restrict__ acc) {
  const int b   = blockIdx.y;
  const int tid = threadIdx.x;
  __shared__ float s_conf[2][MAIN_BUFW];
  __shared__ float s_t[NOBJ][4];
  __shared__ int   s_lab[NOBJ];
  __shared__ unsigned int s_bp[NOBJ];
  __shared__ float s_red[8];

  if (tid < NOBJ * 4) ((float*)s_t)[tid] = tboxes[b * NOBJ * 4 + tid];
  if (tid < NOBJ) {
    s_lab[tid] = tlabels[b * NOBJ + tid];   // raw labels 1..C-1 == labels0+1
    s_bp[tid]  = 0xFFFFFFFFu -
                 (unsigned int)(best_prior[b * NOBJ + tid] & 0xFFFFFFFFull);
  }

  const char*  confB    = (const char*)conf;
  const size_t img_base = (size_t)b * PP * (CC * 4);
  const size_t total    = (size_t)BB * PP * (CC * 4);
  const unsigned ldsA = (unsigned)(size_t)&s_conf[0][0];
  const unsigned ldsB = (unsigned)(size_t)&s_conf[1][0];

  int ti = blockIdx.x;
  // prologue: start streaming the first tile
  asyncCopyB128<MAIN_PT>(ldsA, confB, img_base + (size_t)ti * TILE_BYTES, total);
  __syncthreads();   // covers s_t/s_lab/s_bp visibility too

  float part_sl1 = 0.0f, part_cepos = 0.0f;
  int   cnt_pos  = 0;
  int   par      = 0;

  for (; ti < NT; ti += 4) {
    int  tn      = ti + 4;
    bool hasNext = tn < NT;               // block-uniform
    if (hasNext) {
      asyncCopyB128<MAIN_PT>(par ? ldsA : ldsB, confB,
                             img_base + (size_t)tn * TILE_BYTES, total);
      __builtin_prefetch(&((const float4*)loc)[(size_t)b * PP + tn * 256 + tid], 0, 0);
      S_WAIT_ASYNC(6);                    // current tile resident (in-order)
    } else {
      S_WAIT_ASYNC(0);
    }
    __syncthreads();                      // cross-wave LDS visibility

    const float* cbuf = s_conf[par];
    const int p = ti * 256 + tid;
    if (p < PP) {
      float4 db = ((const float4*)dbox)[p];
      float cxd = db.x, cyd = db.y, wd = db.z, hd = db.w;
      float bx0 = cxd - 0.5f * wd, by0 = cyd - 0.5f * hd;
      float bx1 = cxd + 0.5f * wd, by1 = cyd + 0.5f * hd;
      float ab = wd * hd;

      // best truth for this prior (first max wins, matching argmax axis=0)
      float best = -1.0f;
      int   bidx = 0;
#pragma unroll
      for (int j = 0; j < NOBJ; ++j) {
        float ta = (s_t[j][2] - s_t[j][0]) * (s_t[j][3] - s_t[j][1]);
        float ix0 = fmaxf(s_t[j][0], bx0);
        float iy0 = fmaxf(s_t[j][1], by0);
        float ix1 = fminf(s_t[j][2], bx1);
        float iy1 = fminf(s_t[j][3], by1);
        float iw = fmaxf(ix1 - ix0, 0.0f);
        float ih = fmaxf(iy1 - iy0, 0.0f);
        float inter = iw * ih;
        float iou = inter / (ta + ab - inter);
        if (iou > best) { best = iou; bidx = j; }
      }
      // per-truth best-prior override (ascending j => last write wins,
      // matching the sequential .at[].set semantics)
#pragma unroll
      for (int j = 0; j < NOBJ; ++j) {
        if (s_bp[j] == (unsigned int)p) { best = 2.0f; bidx = j; }
      }
      int  conft = (best < JTHRESH) ? 0 : s_lab[bidx];
      bool pos   = conft > 0;

      // localization target + smooth L1
      float tx0 = s_t[bidx][0], ty0 = s_t[bidx][1];
      float tx1 = s_t[bidx][2], ty1 = s_t[bidx][3];
      float g[4];
      g[0] = (0.5f * (tx0 + tx1) - cxd) / (VAR0 * wd);
      g[1] = (0.5f * (ty0 + ty1) - cyd) / (VAR0 * hd);
      g[2] = __logf((tx1 - tx0) / wd) * (1.0f / VAR1);
      g[3] = __logf((ty1 - ty0) / hd) * (1.0f / VAR1);
      float4 lv = ((const float4*)loc)[(size_t)b * PP + p];
      float l[4] = {lv.x, lv.y, lv.z, lv.w};
      float sl = 0.0f;
#pragma unroll
      for (int i = 0; i < 4; ++i) {
        float d  = l[i] - g[i];
        float ad = fabsf(d);
        sl += (ad < 1.0f) ? 0.5f * d * d : ad - 0.5f;
      }
      if (pos) { part_sl1 += sl; ++cnt_pos; }

      // smoothed CE from the LDS-staged logits row:
      // ce = lse - (1-eps)*x_t - eps*mean(x)
      float x[CC];
      float mx = -3.4e38f;
#pragma unroll
      for (int i = 0; i < CC; ++i) {
        x[i] = cbuf[tid * CC + i];
        mx = fmaxf(mx, x[i]);
      }
      float se = 0.0f, sx = 0.0f, xt = 0.0f;
#pragma unroll
      for (int i = 0; i < CC; ++i) {
        se += __expf(x[i] - mx);
        sx += x[i];
        if (i == conft) xt = x[i];
      }
      float lse = mx + __logf(se);
      float ce  = lse - (1.0f - EPS_LS) * xt - EPS_LS * (sx * (1.0f / CC));

      ce_rank[(size_t)b * PP + p] = pos ? 0.0f : fmaxf(ce, 0.0f);
      if (pos) part_cepos += ce;
    }
    __syncthreads();   // reads done before this buffer is overwritten
    par ^= 1;
  }

  // all threads reconverged: WMMA block reductions (EXEC all-ones)
  float tot_sl1 = blockReduceF32(part_sl1, s_red);
  float tot_cep = blockReduceF32(part_cepos, s_red);
  float tot_np  = blockReduceF32((float)cnt_pos, s_red);
  if (tid == 0) {
    atomicAdd(&acc[0], tot_sl1);
    atomicAdd(&acc[1], tot_cep);
    atomicAdd(&num_pos[b], (int)(tot_np + 0.5f));
  }
}

// ---------------------------------------------------------------------------
// Kernel 3: hard-negative mining. The rank row is staged once into LDS via
// async b128, then 32 radix-select passes + the final sum run LDS-resident.
// ---------------------------------------------------------------------------
__global__ void k_select(const float* __restrict__ ce_rank,
                         const int*   __restrict__ num_pos,
                         float* __restrict__ acc) {
  const int b   = blockIdx.x;
  const int tid = threadIdx.x;
  __shared__ float s_row[SEL_BUFW];
  __shared__ float s_red[8];
  __shared__ unsigned int s_prefix;

  const size_t row_base = (size_t)b * PP * 4;
  const size_t total    = (size_t)BB * PP * 4;
  asyncCopyB128<SEL_PT>((unsigned)(size_t)&s_row[0],
                        (const char*)ce_rank, row_base, total);

  int np = num_pos[b];
  int k  = np * 3;
  if (k > 64) k = 64;
  if (np == 0) k = 32;
  if (k > PP) k = PP;

  if (tid == 0) s_prefix = 0u;
  S_WAIT_ASYNC(0);
  __syncthreads();

  for (int bit = 31; bit >= 0; --bit) {
    unsigned int pf   = s_prefix;
    unsigned int cand = pf | (1u << bit);
    unsigned int hm   = ~((1u << bit) - 1u);
    float cnt = 0.0f;
    for (int p = tid; p < PP; p += blockDim.x) {
      unsigned int v = __float_as_uint(s_row[p]);
      cnt += ((v & hm) >= cand) ? 1.0f : 0.0f;
    }
    float tot = blockReduceF32(cnt, s_red);
    if (tid == 0 && tot >= (float)k) s_prefix = cand;
    __syncthreads();
  }

  unsigned int pf = s_prefix;
  float cgt = 0.0f, sgt = 0.0f;
  for (int p = tid; p < PP; p += blockDim.x) {
    float v = s_row[p];
    if (__float_as_uint(v) > pf) { cgt += 1.0f; sgt += v; }
  }
  float tot_cgt = blockReduceF32(cgt, s_red);
  float tot_sgt = blockReduceF32(sgt, s_red);
  if (tid == 0) {
    float rem = (float)k - tot_cgt;                 // ties taken at threshold
    float thr = __uint_as_float(pf);
    atomicAdd(&acc[2], tot_sgt + rem * thr);
    atomicAdd(&acc[3], (float)k);                   // selected-neg count
  }
}

// ---------------------------------------------------------------------------
// Kernel 4: finalize the two scalars
// ---------------------------------------------------------------------------
__global__ void k_final(const int* __restrict__ num_pos,
                        const float* __restrict__ acc,
                        float* __restrict__ out) {
  __shared__ int s_np[BB];
  int tid = threadIdx.x;
  if (tid < BB) s_np[tid] = num_pos[tid];
  __syncthreads();
  if (tid == 0) {
    int tp = 0;
    for (int i = 0; i < BB; ++i) tp += s_np[i];
    float npf   = (float)(tp > 0 ? tp : 1);
    float lossl = acc[0] / npf;
    float lc    = acc[1] + acc[2];
    float sel   = acc[3] > 0.0f ? acc[3] : 1.0f;
    out[0] = lossl;
    out[1] = (tp > 0) ? lc / npf : lc / sel;
  }
}

// ---------------------------------------------------------------------------
extern "C" void kernel_launch(void* const* d_in, const int* in_sizes, int n_in,
                              void* d_out, int out_size, void* d_ws, size_t ws_size,
                              hipStream_t stream) {
  const float* loc     = (const float*)d_in[0];   // [B,P,4]
  const float* conf    = (const float*)d_in[1];   // [B,P,C]
  const float* dbox    = (const float*)d_in[2];   // [P,4]
  const float* tboxes  = (const float*)d_in[3];   // [B,N,4]
  const int*   tlabels = (const int*)d_in[4];     // [B,N]
  float* out = (float*)d_out;

  char* ws = (char*)d_ws;
  float* ce_rank = (float*)ws;                                  // B*P floats
  size_t off = (size_t)BB * PP * sizeof(float);
  off = (off + 7) & ~(size_t)7;
  unsigned long long* best_prior = (unsigned long long*)(ws + off);
  off += (size_t)BB * NOBJ * sizeof(unsigned long long);
  int* num_pos = (int*)(ws + off);
  off += (size_t)BB * sizeof(int);
  float* acc = (float*)(ws + off);                              // 4 floats

  k_init<<<1, 128, 0, stream>>>(acc, num_pos);
  k_bestprior<<<BB, 256, 0, stream>>>(dbox, tboxes, best_prior);
  k_main<<<dim3(4, BB), 256, 0, stream>>>(loc, conf, dbox, tboxes, tlabels,
                                          best_prior, ce_rank, num_pos, acc);
  k_select<<<BB, 256, 0, stream>>>(ce_rank, num_pos, acc);
  k_final<<<1, 128, 0, stream>>>(num_pos, acc, out);
}